// GNN_1340029796803
// MI455X (gfx1250) — compile-verified
//
#include <hip/hip_runtime.h>
#include <hip/hip_bf16.h>
#include <math.h>

typedef float v2f __attribute__((ext_vector_type(2)));
typedef float v8f __attribute__((ext_vector_type(8)));

#define HDIM 64

// ---------- helpers: order-preserving float<->uint for atomic max ----------
__device__ __forceinline__ unsigned fenc(float f) {
    unsigned u = __float_as_uint(f);
    return (u >> 31) ? ~u : (u | 0x80000000u);
}
__device__ __forceinline__ float fdec(unsigned u) {
    return (u >> 31) ? __uint_as_float(u & 0x7fffffffu) : __uint_as_float(~u);
}

// ---------- prep: u1 = We1@ae1 ; u2 = We1@(We2@ae2) ; u3 = We1@(We2@(We3@ae3)) ----------
// one block, 64 threads. u layout: u[0..31]=u1, u[32..63]=u2, u[64..95]=u3
__global__ __launch_bounds__(64) void prep_u_kernel(
    const float* __restrict__ We1, const float* __restrict__ We2, const float* __restrict__ We3,
    const float* __restrict__ ae1, const float* __restrict__ ae2, const float* __restrict__ ae3,
    float* __restrict__ u)
{
    __shared__ float sa[64], sb[64];
    int k = threadIdx.x;
    float acc;

    // t3 = We3 @ ae3
    acc = 0.f;
    for (int h = 0; h < 64; ++h) acc += We3[k * 64 + h] * ae3[h];
    sa[k] = acc; __syncthreads();
    // t23 = We2 @ t3
    acc = 0.f;
    for (int h = 0; h < 64; ++h) acc += We2[k * 64 + h] * sa[h];
    sb[k] = acc; __syncthreads();
    // u3 = We1 @ t23
    if (k < 32) {
        acc = 0.f;
        for (int h = 0; h < 64; ++h) acc += We1[k * 64 + h] * sb[h];
        u[64 + k] = acc;
    }
    __syncthreads();
    // t2 = We2 @ ae2
    acc = 0.f;
    for (int h = 0; h < 64; ++h) acc += We2[k * 64 + h] * ae2[h];
    sa[k] = acc; __syncthreads();
    if (k < 32) {
        // u2 = We1 @ t2
        acc = 0.f;
        for (int h = 0; h < 64; ++h) acc += We1[k * 64 + h] * sa[h];
        u[32 + k] = acc;
        // u1 = We1 @ ae1
        float a1 = 0.f;
        for (int h = 0; h < 64; ++h) a1 += We1[k * 64 + h] * ae1[h];
        u[k] = a1;
    }
}

// ---------- edge dots: one pass over edge_attr, produce s_e for all 3 layers ----------
__global__ __launch_bounds__(256) void edge_dots_kernel(
    const float* __restrict__ eattr, const float* __restrict__ u,
    float* __restrict__ se1, float* __restrict__ se2, float* __restrict__ se3, int E)
{
    __shared__ float su[96];
    if (threadIdx.x < 96) su[threadIdx.x] = u[threadIdx.x];
    __syncthreads();
    int e = blockIdx.x * 256 + threadIdx.x;
    if (e >= E) return;
    const float4* row = (const float4*)(eattr + (size_t)e * 32);
    float d1 = 0.f, d2 = 0.f, d3 = 0.f;
#pragma unroll
    for (int q = 0; q < 8; ++q) {
        float4 v = row[q];
        int b = q * 4;
        d1 += v.x * su[b] + v.y * su[b + 1] + v.z * su[b + 2] + v.w * su[b + 3];
        d2 += v.x * su[32 + b] + v.y * su[33 + b] + v.z * su[34 + b] + v.w * su[35 + b];
        d3 += v.x * su[64 + b] + v.y * su[65 + b] + v.z * su[66 + b] + v.w * su[67 + b];
    }
    se1[e] = d1; se2[e] = d2; se3[e] = d3;
}

// ---------- node projection GEMM via V_WMMA_F32_16X16X4_F32, fused score dots ----------
// XP[nrows x 64] = X[nrows x K] @ W[K x 64];  s_src = XP@a_s ; s_dst = XP@a_d
// one wave computes a 16x64 row tile. 8 waves / block. nrows must be multiple of 16.
__global__ __launch_bounds__(256) void node_gemm_kernel(
    const float* __restrict__ X, const float* __restrict__ W,
    float* __restrict__ XP,
    const float* __restrict__ a_s, const float* __restrict__ a_d,
    float* __restrict__ s_src, float* __restrict__ s_dst,
    int nrows, int K)
{
    int lane = threadIdx.x & 31;
    int wv = threadIdx.x >> 5;
    int rowBase = (blockIdx.x * 8 + wv) * 16;
    if (rowBase >= nrows) return;           // wave-uniform: EXEC stays all-ones for WMMA

    int m  = lane & 15;                     // row (A) / col (B,C) within tile
    int hi = lane >> 4;                     // lane-half selects K pair / row half

    v8f c0 = {}, c1 = {}, c2 = {}, c3 = {};
    const float* xrow = X + (size_t)(rowBase + m) * K;
    int nkt = K >> 2;

    for (int kt = 0; kt < nkt; ++kt) {
        int k0 = kt * 4 + hi * 2;
        // A 16x4 fp32: lane m holds row m; vgpr0 = K=k0, vgpr1 = K=k0+1 (halves offset by 2)
        v2f a;
        a.x = xrow[k0];
        a.y = xrow[k0 + 1];
        // B 4x16 fp32: lane m holds col m; per-lane pair covers K=k0,k0+1
        const float* w0 = W + (size_t)k0 * 64 + m;
        const float* w1 = w0 + 64;
        v2f b0, b1, b2, b3;
        b0.x = w0[0];  b0.y = w1[0];
        b1.x = w0[16]; b1.y = w1[16];
        b2.x = w0[32]; b2.y = w1[32];
        b3.x = w0[48]; b3.y = w1[48];
        c0 = __builtin_amdgcn_wmma_f32_16x16x4_f32(false, a, false, b0, (short)0, c0, false, false);
        c1 = __builtin_amdgcn_wmma_f32_16x16x4_f32(false, a, false, b1, (short)0, c1, false, false);
        c2 = __builtin_amdgcn_wmma_f32_16x16x4_f32(false, a, false, b2, (short)0, c2, false, false);
        c3 = __builtin_amdgcn_wmma_f32_16x16x4_f32(false, a, false, b3, (short)0, c3, false, false);
    }

    // attention vectors for this lane's column in each 16-wide N tile
    float as0 = a_s[m], as1 = a_s[16 + m], as2 = a_s[32 + m], as3 = a_s[48 + m];
    float ad0 = a_d[m], ad1 = a_d[16 + m], ad2 = a_d[32 + m], ad3 = a_d[48 + m];

#pragma unroll
    for (int j = 0; j < 8; ++j) {
        int r = rowBase + hi * 8 + j;       // C layout: vgpr j <-> row j + 8*hi, col = m
        float* orow = XP + (size_t)r * 64;
        orow[m]      = c0[j];
        orow[16 + m] = c1[j];
        orow[32 + m] = c2[j];
        orow[48 + m] = c3[j];
        float ps = c0[j] * as0 + c1[j] * as1 + c2[j] * as2 + c3[j] * as3;
        float pd = c0[j] * ad0 + c1[j] * ad1 + c2[j] * ad2 + c3[j] * ad3;
#pragma unroll
        for (int off = 8; off; off >>= 1) { // reduce over the 16 lanes sharing this row
            ps += __shfl_xor(ps, off, 16);
            pd += __shfl_xor(pd, off, 16);
        }
        if (m == 0) { s_src[r] = ps; s_dst[r] = pd; }
    }
}

// ---------- per-layer init ----------
__global__ __launch_bounds__(256) void init_kernel(
    float* __restrict__ xnext, float* __restrict__ denom, unsigned* __restrict__ menc, int N)
{
    int id = blockIdx.x * 256 + threadIdx.x;
    if (id < N * HDIM) xnext[id] = 0.f;
    if (id < N) { denom[id] = 0.f; menc[id] = 0u; }
}

// ---------- edge scores + segment max ----------
__global__ __launch_bounds__(256) void score_kernel(
    const int* __restrict__ src, const int* __restrict__ dst,
    const float* __restrict__ s_src, const float* __restrict__ s_dst,
    const float* __restrict__ s_e,
    float* __restrict__ score, unsigned* __restrict__ menc, int E)
{
    int e = blockIdx.x * 256 + threadIdx.x;
    if (e >= E) return;
    int s = src[e], d = dst[e];
    float sc = s_src[s] + s_dst[d] + s_e[e];
    sc = (sc > 0.f) ? sc : 0.2f * sc;       // leaky relu
    score[e] = sc;
    atomicMax(&menc[d], fenc(sc));
}

// ---------- exp + unnormalized scatter-accumulate (one wave per edge) ----------
__global__ __launch_bounds__(256) void accum_kernel(
    const int* __restrict__ src, const int* __restrict__ dst,
    const float* __restrict__ score, const unsigned* __restrict__ menc,
    float* __restrict__ denom, const float* __restrict__ xp,
    float* __restrict__ xnext, int E)
{
    int lane = threadIdx.x & 31;
    int e = blockIdx.x * 8 + (threadIdx.x >> 5);
    if (e >= E) return;
    int s = src[e], d = dst[e];
    float ex = __expf(score[e] - fdec(menc[d]));
    if (lane == 0) atomicAdd(&denom[d], ex);
    const float2* xr = (const float2*)(xp + (size_t)s * 64);
    float2 v = xr[lane];
    float* o = xnext + (size_t)d * 64 + lane * 2;
    atomicAdd(o,     ex * v.x);
    atomicAdd(o + 1, ex * v.y);
}

// ---------- finalize: divide by denom, add bias, relu ----------
__global__ __launch_bounds__(256) void finalize_kernel(
    const float* __restrict__ xacc, const float* __restrict__ denom,
    const float* __restrict__ b, float* __restrict__ xout, int N)
{
    int id = blockIdx.x * 256 + threadIdx.x;
    if (id >= N * HDIM) return;
    int i = id >> 6, h = id & 63;
    float v = xacc[id] / (denom[i] + 1e-16f) + b[h];
    xout[id] = fmaxf(v, 0.f);
}

// ---------- final projection to [N,1] ----------
__global__ __launch_bounds__(256) void final_dot_kernel(
    const float* __restrict__ x3, const float* __restrict__ Wc,
    const float* __restrict__ bc, float* __restrict__ out, int N)
{
    __shared__ float wc[64];
    if (threadIdx.x < 64) wc[threadIdx.x] = Wc[threadIdx.x];
    __syncthreads();
    int i = blockIdx.x * 256 + threadIdx.x;
    if (i >= N) return;
    const float4* r = (const float4*)(x3 + (size_t)i * 64);
    float acc = 0.f;
#pragma unroll
    for (int q = 0; q < 16; ++q) {
        float4 v = r[q];
        acc += v.x * wc[q * 4] + v.y * wc[q * 4 + 1] + v.z * wc[q * 4 + 2] + v.w * wc[q * 4 + 3];
    }
    out[i] = acc + bc[0];
}

extern "C" void kernel_launch(void* const* d_in, const int* in_sizes, int n_in,
                              void* d_out, int out_size, void* d_ws, size_t ws_size,
                              hipStream_t stream) {
    const int N = in_sizes[0] / 128;
    const int E = in_sizes[2] / 32;

    const float* x     = (const float*)d_in[0];
    const int*   ei    = (const int*)d_in[1];
    const int*   src   = ei;
    const int*   dst   = ei + E;
    const float* eattr = (const float*)d_in[2];

    const float* W[3]  = { (const float*)d_in[3],  (const float*)d_in[9],  (const float*)d_in[15] };
    const float* We[3] = { (const float*)d_in[4],  (const float*)d_in[10], (const float*)d_in[16] };
    const float* as[3] = { (const float*)d_in[5],  (const float*)d_in[11], (const float*)d_in[17] };
    const float* ad[3] = { (const float*)d_in[6],  (const float*)d_in[12], (const float*)d_in[18] };
    const float* ae[3] = { (const float*)d_in[7],  (const float*)d_in[13], (const float*)d_in[19] };
    const float* b[3]  = { (const float*)d_in[8],  (const float*)d_in[14], (const float*)d_in[20] };
    const float* Wc    = (const float*)d_in[21];
    const float* bc    = (const float*)d_in[22];

    // workspace layout (floats)
    float* p = (float*)d_ws;
    float* u     = p; p += 128;
    float* se[3]; se[0] = p; p += E; se[1] = p; p += E; se[2] = p; p += E;
    float* ssrc  = p; p += N;
    float* sdst  = p; p += N;
    float* score = p; p += E;
    float* denom = p; p += N;
    unsigned* menc = (unsigned*)p; p += N;
    float* xp = p; p += (size_t)N * HDIM;
    float* B0 = p; p += (size_t)N * HDIM;
    float* B1 = p; p += (size_t)N * HDIM;

    const int gridE   = (E + 255) / 256;
    const int gridNH  = (N * HDIM + 255) / 256;
    const int gridN   = (N + 255) / 256;
    const int gridGem = (((N + 15) / 16) + 7) / 8;
    const int gridAcc = (E + 7) / 8;

    // fold the whole edge-feature pathway into three E-length score vectors
    prep_u_kernel<<<1, 64, 0, stream>>>(We[0], We[1], We[2], ae[0], ae[1], ae[2], u);
    edge_dots_kernel<<<gridE, 256, 0, stream>>>(eattr, u, se[0], se[1], se[2], E);

    const float* xcur = x;
    float* xnext[3] = { B0, B1, B0 };
    int Kl[3] = { 128, HDIM, HDIM };

    for (int l = 0; l < 3; ++l) {
        node_gemm_kernel<<<gridGem, 256, 0, stream>>>(xcur, W[l], xp, as[l], ad[l],
                                                      ssrc, sdst, N, Kl[l]);
        init_kernel<<<gridNH, 256, 0, stream>>>(xnext[l], denom, menc, N);
        score_kernel<<<gridE, 256, 0, stream>>>(src, dst, ssrc, sdst, se[l], score, menc, E);
        accum_kernel<<<gridAcc, 256, 0, stream>>>(src, dst, score, menc, denom, xp, xnext[l], E);
        finalize_kernel<<<gridNH, 256, 0, stream>>>(xnext[l], denom, b[l], xnext[l], N);
        xcur = xnext[l];
    }

    final_dot_kernel<<<gridN, 256, 0, stream>>>(B0, Wc, bc, (float*)d_out, N);
}